// MyLayer_37125697307424
// MI455X (gfx1250) — compile-verified
//
#include <hip/hip_runtime.h>
#include <math.h>

typedef __attribute__((ext_vector_type(2))) float v2f;
typedef __attribute__((ext_vector_type(8))) float v8f;

#define HEADS 4
#define DDIM  32
#define FN    128
#define FE    64
#define NOUTC 128   // H*DN
#define SLOPE 0.2f

// ---- order-preserving float<->uint mapping for atomicMax on floats ----
__device__ __forceinline__ unsigned f2ord(float f) {
  unsigned u = __float_as_uint(f);
  return (u & 0x80000000u) ? ~u : (u | 0x80000000u);
}
__device__ __forceinline__ float ord2f(unsigned u) {
  u = (u & 0x80000000u) ? (u & 0x7fffffffu) : ~u;
  return __uint_as_float(u);
}

// ---------------------------------------------------------------------
// C(M,Ncols) = A(M,K) @ B(K,Ncols) + bias, all fp32, one wave per 16x16
// tile, K stepped by 4 using V_WMMA_F32_16X16X4_F32.
// Requires M % 16 == 0, K % 4 == 0, Ncols % 16 == 0 (true for this net).
// ---------------------------------------------------------------------
__global__ void gemm16_wmma(const float* __restrict__ A, const float* __restrict__ B,
                            const float* __restrict__ bias, float* __restrict__ C,
                            int M, int K, int Ncols) {
  int wid  = (blockIdx.x * blockDim.x + threadIdx.x) >> 5;
  int lane = threadIdx.x & 31;
  int tilesN = Ncols >> 4;
  int tm = (wid / tilesN) << 4;
  int tn = (wid % tilesN) << 4;
  if (tm >= M) return;                 // wave-uniform exit, EXEC stays full
  int half = lane >> 4;                // 0: K pair {0,1}, 1: K pair {2,3}
  int l16  = lane & 15;
  const float* arow = A + (size_t)(tm + l16) * K;
  v8f acc = {};
  for (int k = 0; k < K; k += 4) {
    int ka = k + (half << 1);
    v2f a, b;
    a.x = arow[ka];
    a.y = arow[ka + 1];
    b.x = B[(size_t)ka       * Ncols + tn + l16];
    b.y = B[(size_t)(ka + 1) * Ncols + tn + l16];
    acc = __builtin_amdgcn_wmma_f32_16x16x4_f32(false, a, false, b,
                                                (short)0, acc, false, false);
  }
  float bv = bias[tn + l16];
#pragma unroll
  for (int i = 0; i < 8; i++) {
    int row = tm + i + (half << 3);    // D: VGPR i -> rows i / i+8
    C[(size_t)row * Ncols + tn + l16] = acc[i] + bv;
  }
}

// new_edge_feats = relu(feat_e).mean(heads); one thread per (edge, d)
__global__ void edge_out_kernel(const float* __restrict__ feat_e,
                                float* __restrict__ out_e, int E) {
  int idx = blockIdx.x * blockDim.x + threadIdx.x;
  if (idx >= E * DDIM) return;
  int e = idx >> 5, d = idx & 31;
  const float* fe = feat_e + (size_t)e * NOUTC;
  float s = 0.f;
#pragma unroll
  for (int h = 0; h < HEADS; h++) s += fmaxf(fe[h * DDIM + d], 0.f);
  out_e[idx] = s * 0.25f;
}

// per-edge attention logits (both GATs) + atomicMax per (dst, head)
__global__ void edge_logits_kernel(const float* __restrict__ feat_src,
                                   const float* __restrict__ feat_e,
                                   const int* __restrict__ src, const int* __restrict__ dst,
                                   const float* __restrict__ attn_n2n,
                                   const float* __restrict__ attn_e2n,
                                   float* __restrict__ log_n2n, float* __restrict__ log_e2n,
                                   unsigned* __restrict__ max_n2n,
                                   unsigned* __restrict__ max_e2n, int E) {
  int wid  = (blockIdx.x * blockDim.x + threadIdx.x) >> 5;
  int lane = threadIdx.x & 31;
  if (wid >= E) return;
  size_t e = (size_t)wid;
  int s = src[e], d = dst[e];
#pragma unroll
  for (int h = 0; h < HEADS; h++) {
    float el = feat_src[(size_t)s * NOUTC + h * DDIM + lane];
    float fd = feat_src[(size_t)d * NOUTC + h * DDIM + lane];
    float fe = feat_e[e * NOUTC + h * DDIM + lane];
    float p1 = (el + fd) * attn_n2n[h * DDIM + lane];
    float p2 = (el + fe) * attn_e2n[h * DDIM + lane];
#pragma unroll
    for (int off = 16; off > 0; off >>= 1) {
      p1 += __shfl_down(p1, off, 32);
      p2 += __shfl_down(p2, off, 32);
    }
    if (lane == 0) {
      float l1 = p1 > 0.f ? p1 : p1 * SLOPE;
      float l2 = p2 > 0.f ? p2 : p2 * SLOPE;
      log_n2n[e * HEADS + h] = l1;
      log_e2n[e * HEADS + h] = l2;
      atomicMax(&max_n2n[(size_t)d * HEADS + h], f2ord(l1));
      atomicMax(&max_e2n[(size_t)d * HEADS + h], f2ord(l2));
    }
  }
}

// ex = exp(logit - max[dst]); accumulate segment sums; one thread per (e,h)
__global__ void expsum_kernel(const int* __restrict__ dst,
                              float* __restrict__ ex_n2n, float* __restrict__ ex_e2n,
                              const unsigned* __restrict__ max_n2n,
                              const unsigned* __restrict__ max_e2n,
                              float* __restrict__ sum_n2n, float* __restrict__ sum_e2n,
                              int E) {
  int idx = blockIdx.x * blockDim.x + threadIdx.x;
  if (idx >= E * HEADS) return;
  int e = idx >> 2, h = idx & 3;
  int d = dst[e];
  float x1 = __expf(ex_n2n[idx] - ord2f(max_n2n[(size_t)d * HEADS + h]));
  float x2 = __expf(ex_e2n[idx] - ord2f(max_e2n[(size_t)d * HEADS + h]));
  ex_n2n[idx] = x1;
  ex_e2n[idx] = x2;
  atomicAdd(&sum_n2n[(size_t)d * HEADS + h], x1);
  atomicAdd(&sum_e2n[(size_t)d * HEADS + h], x2);
}

// weighted scatter: node_node += a_n2n*el ; edge_node += a_e2n*feat_e
__global__ void aggregate_kernel(const float* __restrict__ feat_src,
                                 const float* __restrict__ feat_e,
                                 const int* __restrict__ src, const int* __restrict__ dst,
                                 const float* __restrict__ ex_n2n, const float* __restrict__ ex_e2n,
                                 const float* __restrict__ sum_n2n, const float* __restrict__ sum_e2n,
                                 float* __restrict__ node_node, float* __restrict__ edge_node,
                                 int E) {
  int wid  = (blockIdx.x * blockDim.x + threadIdx.x) >> 5;
  int lane = threadIdx.x & 31;
  if (wid >= E) return;
  size_t e = (size_t)wid;
  int s = src[e], d = dst[e];
#pragma unroll
  for (int h = 0; h < HEADS; h++) {
    float a1 = ex_n2n[e * HEADS + h] / sum_n2n[(size_t)d * HEADS + h];
    float a2 = ex_e2n[e * HEADS + h] / sum_e2n[(size_t)d * HEADS + h];
    float el = feat_src[(size_t)s * NOUTC + h * DDIM + lane];
    float fe = feat_e[e * NOUTC + h * DDIM + lane];
    atomicAdd(&node_node[((size_t)d * HEADS + h) * DDIM + lane], el * a1);
    atomicAdd(&edge_node[((size_t)d * HEADS + h) * DDIM + lane], fe * a2);
  }
}

// per-node: relu -> concat@Wu+bu -> relu -> head mean -> GRU step
// one wave per node, lane = feature index, weights staged in LDS
__global__ void node_update_kernel(const float* __restrict__ feat_src,
                                   const float* __restrict__ node_node,
                                   const float* __restrict__ edge_node,
                                   const float* __restrict__ Wu, const float* __restrict__ bu,
                                   const float* __restrict__ W_ih, const float* __restrict__ W_hh,
                                   const float* __restrict__ b_ih, const float* __restrict__ b_hh,
                                   float* __restrict__ out, int N) {
  __shared__ float sWu[64 * 32];
  __shared__ float sWih[96 * 32];
  __shared__ float sWhh[96 * 32];
  for (int i = threadIdx.x; i < 64 * 32; i += blockDim.x) sWu[i] = Wu[i];
  for (int i = threadIdx.x; i < 96 * 32; i += blockDim.x) { sWih[i] = W_ih[i]; sWhh[i] = W_hh[i]; }
  __syncthreads();

  int wid  = (blockIdx.x * blockDim.x + threadIdx.x) >> 5;
  int lane = threadIdx.x & 31;
  if (wid >= N) return;
  size_t n = (size_t)wid;

  float bub = bu[lane];
  float x = 0.f;
#pragma unroll
  for (int h = 0; h < HEADS; h++) {
    float nnv = fmaxf(node_node[(n * HEADS + h) * DDIM + lane], 0.f);
    float env = fmaxf(edge_node[(n * HEADS + h) * DDIM + lane], 0.f);
    float acc = bub;
#pragma unroll
    for (int d = 0; d < 32; d++)
      acc += __shfl(nnv, d, 32) * sWu[d * 32 + lane];
#pragma unroll
    for (int d = 0; d < 32; d++)
      acc += __shfl(env, d, 32) * sWu[(32 + d) * 32 + lane];
    x += fmaxf(acc, 0.f);
  }
  x *= 0.25f;

  float h0 = 0.f;
#pragma unroll
  for (int h = 0; h < HEADS; h++) h0 += feat_src[n * NOUTC + h * DDIM + lane];
  h0 *= 0.25f;

  float gi[3], gh[3];
#pragma unroll
  for (int g = 0; g < 3; g++) {
    float a = b_ih[g * 32 + lane];
    float b = b_hh[g * 32 + lane];
#pragma unroll
    for (int o = 0; o < 32; o++) {
      a += __shfl(x, o, 32)  * sWih[(g * 32 + lane) * 32 + o];
      b += __shfl(h0, o, 32) * sWhh[(g * 32 + lane) * 32 + o];
    }
    gi[g] = a; gh[g] = b;
  }
  float r  = 1.f / (1.f + __expf(-(gi[0] + gh[0])));
  float z  = 1.f / (1.f + __expf(-(gi[1] + gh[1])));
  float ng = tanhf(gi[2] + r * gh[2]);
  out[n * DDIM + lane] = (1.f - z) * ng + z * h0;
}

extern "C" void kernel_launch(void* const* d_in, const int* in_sizes, int n_in,
                              void* d_out, int out_size, void* d_ws, size_t ws_size,
                              hipStream_t stream) {
  const float* node_feats = (const float*)d_in[0];
  const float* edge_feats = (const float*)d_in[1];
  const int*   src        = (const int*)d_in[2];
  const int*   dst        = (const int*)d_in[3];
  const float* Wn         = (const float*)d_in[4];
  const float* bn         = (const float*)d_in[5];
  const float* attn_n2n   = (const float*)d_in[6];
  const float* We         = (const float*)d_in[7];
  const float* be         = (const float*)d_in[8];
  const float* attn_e2n   = (const float*)d_in[9];
  const float* Wu         = (const float*)d_in[10];
  const float* bu         = (const float*)d_in[11];
  const float* W_ih       = (const float*)d_in[12];
  const float* W_hh       = (const float*)d_in[13];
  const float* b_ih       = (const float*)d_in[14];
  const float* b_hh       = (const float*)d_in[15];

  const int N = in_sizes[0] / FN;    // 50000
  const int E = in_sizes[1] / FE;    // 800000

  float* out_nodes = (float*)d_out;                       // (N, 32)
  float* out_edges = out_nodes + (size_t)N * DDIM;        // (E, 32)

  // ---- workspace partition (256B aligned) ----
  size_t off = 0;
  auto take = [&](size_t elems) -> char* {
    char* p = (char*)d_ws + off;
    off += (elems * sizeof(float) + 255) & ~(size_t)255;
    return p;
  };
  float*    feat_src = (float*)take((size_t)N * NOUTC);
  float*    feat_e   = (float*)take((size_t)E * NOUTC);
  float*    ex_n2n   = (float*)take((size_t)E * HEADS);   // logits, then exp()
  float*    ex_e2n   = (float*)take((size_t)E * HEADS);
  size_t zero_begin = off;
  unsigned* max_n2n  = (unsigned*)take((size_t)N * HEADS);
  unsigned* max_e2n  = (unsigned*)take((size_t)N * HEADS);
  float*    sum_n2n  = (float*)take((size_t)N * HEADS);
  float*    sum_e2n  = (float*)take((size_t)N * HEADS);
  float*    node_nd  = (float*)take((size_t)N * NOUTC);
  float*    edge_nd  = (float*)take((size_t)N * NOUTC);
  size_t zero_end = off;

  hipMemsetAsync((char*)d_ws + zero_begin, 0, zero_end - zero_begin, stream);

  // 1) projections (WMMA fp32)
  {
    int tiles = (N / 16) * (NOUTC / 16);
    gemm16_wmma<<<(tiles + 7) / 8, 256, 0, stream>>>(node_feats, Wn, bn, feat_src, N, FN, NOUTC);
  }
  {
    int tiles = (E / 16) * (NOUTC / 16);
    gemm16_wmma<<<(tiles + 7) / 8, 256, 0, stream>>>(edge_feats, We, be, feat_e, E, FE, NOUTC);
  }
  // 2) edge output (independent of attention)
  edge_out_kernel<<<((E * DDIM) + 255) / 256, 256, 0, stream>>>(feat_e, out_edges, E);
  // 3) logits + segment max
  edge_logits_kernel<<<(E + 7) / 8, 256, 0, stream>>>(feat_src, feat_e, src, dst,
                                                      attn_n2n, attn_e2n,
                                                      ex_n2n, ex_e2n, max_n2n, max_e2n, E);
  // 4) exp + segment sum
  expsum_kernel<<<((E * HEADS) + 255) / 256, 256, 0, stream>>>(dst, ex_n2n, ex_e2n,
                                                               max_n2n, max_e2n,
                                                               sum_n2n, sum_e2n, E);
  // 5) weighted scatter into node accumulators
  aggregate_kernel<<<(E + 7) / 8, 256, 0, stream>>>(feat_src, feat_e, src, dst,
                                                    ex_n2n, ex_e2n, sum_n2n, sum_e2n,
                                                    node_nd, edge_nd, E);
  // 6) node update + GRU
  node_update_kernel<<<(N + 7) / 8, 256, 0, stream>>>(feat_src, node_nd, edge_nd,
                                                      Wu, bu, W_ih, W_hh, b_ih, b_hh,
                                                      out_nodes, N);
}